// SpikeActivation_34969623724323
// MI455X (gfx1250) — compile-verified
//
#include <hip/hip_runtime.h>

// Problem dims (fixed by reference setup_inputs): (B, C, L) = (64, 256, 512)
#define B_DIM   64
#define C_DIM   256
#define L_DIM   512
#define LIF_BETA 0.9f
#define NSTEP   16
#define BN_EPS  1e-5f

// ---------------------------------------------------------------------------
// Kernel A: per-channel batch-norm statistics -> (scale, bias) in workspace.
// One block per channel; 256 threads. Each iteration the block covers TWO
// batch rows with b128 loads: lanes 0..127 -> row b, lanes 128..255 -> row b+1
// (each row = 512 floats = 128 float4s). 32 iterations cover all 64 rows.
// ---------------------------------------------------------------------------
__global__ __launch_bounds__(256) void bn_stats_kernel(
    const float* __restrict__ act,
    const float* __restrict__ gamma,
    const float* __restrict__ beta_bn,
    float* __restrict__ sb)   // sb[0..C-1] = scale, sb[C..2C-1] = bias
{
    const int c = blockIdx.x;
    const int t = threadIdx.x;
    const int lane_l = (t & 127) * 4;     // element offset within a row
    const int bsel   = t >> 7;            // 0 or 1: which of the row pair

    const size_t bstride = (size_t)C_DIM * L_DIM;   // one batch step
    const float* base = act + (size_t)c * L_DIM + (size_t)bsel * bstride + lane_l;

    float s = 0.0f, sq = 0.0f;
#pragma unroll 4
    for (int b = 0; b < B_DIM; b += 2) {
        const float4 v = *(const float4*)(base + (size_t)b * bstride);
        s  += (v.x + v.y) + (v.z + v.w);
        sq  = __builtin_fmaf(v.x, v.x, sq);
        sq  = __builtin_fmaf(v.y, v.y, sq);
        sq  = __builtin_fmaf(v.z, v.z, sq);
        sq  = __builtin_fmaf(v.w, v.w, sq);
    }

    __shared__ float rs[256];
    __shared__ float rq[256];
    rs[t] = s;
    rq[t] = sq;
    __syncthreads();
    for (int off = 128; off > 0; off >>= 1) {
        if (t < off) {
            rs[t] += rs[t + off];
            rq[t] += rq[t + off];
        }
        __syncthreads();
    }

    if (t == 0) {
        const float inv_n = 1.0f / (float)(B_DIM * L_DIM);
        const float mean  = rs[0] * inv_n;
        const float var   = rq[0] * inv_n - mean * mean;   // biased (ddof=0)
        const float scale = gamma[c] / sqrtf(var + BN_EPS);
        sb[c]         = scale;
        sb[C_DIM + c] = beta_bn[c] - mean * scale;
    }
}

// ---------------------------------------------------------------------------
// Kernel B: LIF recurrence (VALU-bound). Each block: 128 threads, TWO (b,c)
// rows of the same channel -> 8 independent LIF chains per thread for deep
// ILP / VOPD dual-issue. Rows are staged global->LDS via the CDNA5 async
// copy path (GLOBAL_LOAD_ASYNC_TO_LDS_B128, ASYNCcnt-tracked). Each lane
// reads back only the bytes it copied, so a per-wave s_wait_asynccnt 0 is
// sufficient (no workgroup barrier needed).
// ---------------------------------------------------------------------------
__global__ __launch_bounds__(128) void lif_kernel(
    const float* __restrict__ act,
    const float* __restrict__ threshold,   // (C,1)
    const float* __restrict__ sb,          // scale/bias from kernel A
    float* __restrict__ out)
{
    const int bid = blockIdx.x;            // bid = bp*C + c, bp in [0,32)
    const int c   = bid & (C_DIM - 1);     // uniform per block -> scalar loads
    const int bp  = bid >> 8;              // batch pair index
    const int t   = threadIdx.x;

    const size_t bstride = (size_t)C_DIM * L_DIM;
    const size_t row0 = (size_t)(2 * bp) * bstride + (size_t)c * L_DIM;
    const size_t row1 = row0 + bstride;

    const float scale = sb[c];
    const float bias  = sb[C_DIM + c];
    const float thr   = threshold[c];

    __shared__ __align__(16) float srow[2 * L_DIM];

    // Async copies: each lane moves 2 x 16 bytes global -> LDS.
    {
        unsigned lds0 = (unsigned)(size_t)(&srow[t * 4]);
        unsigned lds1 = (unsigned)(size_t)(&srow[L_DIM + t * 4]);
        unsigned long long ga0 = (unsigned long long)(act + row0 + t * 4);
        unsigned long long ga1 = (unsigned long long)(act + row1 + t * 4);
        asm volatile("global_load_async_to_lds_b128 %0, %1, off"
                     :: "v"(lds0), "v"(ga0) : "memory");
        asm volatile("global_load_async_to_lds_b128 %0, %1, off"
                     :: "v"(lds1), "v"(ga1) : "memory");
        asm volatile("s_wait_asynccnt 0" ::: "memory");
    }
    const float4 a0 = *(const float4*)(&srow[t * 4]);
    const float4 a1 = *(const float4*)(&srow[L_DIM + t * 4]);

    // BN affine + precompute x - thr for the reset-select. 8 chains.
    float x[8]  = {a0.x, a0.y, a0.z, a0.w, a1.x, a1.y, a1.z, a1.w};
    float xm[8];
    float mem[8];
    float cnt[8];
    bool  r[8];
#pragma unroll
    for (int i = 0; i < 8; ++i) {
        x[i]   = __builtin_fmaf(x[i], scale, bias);
        xm[i]  = x[i] - thr;
        mem[i] = 0.0f;
        cnt[i] = 0.0f;
        r[i]   = (0.0f > thr);             // reset indicator for mem0 = 0
    }

    // 16 LIF steps, 8 independent chains:
    //   mem = beta*mem + (r ? x - thr : x);  spk = mem > thr;  cnt += spk
#pragma unroll
    for (int s = 0; s < NSTEP; ++s) {
#pragma unroll
        for (int i = 0; i < 8; ++i) {
            mem[i] = __builtin_fmaf(LIF_BETA, mem[i], r[i] ? xm[i] : x[i]);
            r[i]   = mem[i] > thr;
            cnt[i] += r[i] ? 1.0f : 0.0f;
        }
    }

    float4 o0, o1;
    o0.x = cnt[0] * (1.0f / NSTEP);
    o0.y = cnt[1] * (1.0f / NSTEP);
    o0.z = cnt[2] * (1.0f / NSTEP);
    o0.w = cnt[3] * (1.0f / NSTEP);
    o1.x = cnt[4] * (1.0f / NSTEP);
    o1.y = cnt[5] * (1.0f / NSTEP);
    o1.z = cnt[6] * (1.0f / NSTEP);
    o1.w = cnt[7] * (1.0f / NSTEP);
    *(float4*)(out + row0 + t * 4) = o0;
    *(float4*)(out + row1 + t * 4) = o1;
}

// ---------------------------------------------------------------------------
extern "C" void kernel_launch(void* const* d_in, const int* in_sizes, int n_in,
                              void* d_out, int out_size, void* d_ws, size_t ws_size,
                              hipStream_t stream)
{
    const float* act      = (const float*)d_in[0];  // (64,256,512) f32
    const float* gamma    = (const float*)d_in[1];  // (256,)
    const float* beta_bn  = (const float*)d_in[2];  // (256,)
    const float* thr      = (const float*)d_in[3];  // (256,1)
    float* out = (float*)d_out;                     // (64,256,512) f32
    float* sb  = (float*)d_ws;                      // 512 floats scratch

    bn_stats_kernel<<<C_DIM, 256, 0, stream>>>(act, gamma, beta_bn, sb);
    lif_kernel<<<(B_DIM / 2) * C_DIM, 128, 0, stream>>>(act, thr, sb, out);
}